// RNN_34282428956800
// MI455X (gfx1250) — compile-verified
//
#include <hip/hip_runtime.h>

// Problem sizes (fixed by the reference)
#define S_LEN 1024
#define B_SZ  512
#define I_DIM 5
#define H_DIM 32
#define G_DIM 128   // 4*H, gate order: i, f, g, o

typedef float v2f __attribute__((ext_vector_type(2)));
typedef float v8f __attribute__((ext_vector_type(8)));

// D = A(16x4 f32) * B(4x16 f32) + C(16x16 f32), one matrix striped across the wave.
__device__ __forceinline__ v8f wmma4(v2f a, v2f b, v8f c) {
    return __builtin_amdgcn_wmma_f32_16x16x4_f32(
        /*neg_a=*/false, a, /*neg_b=*/false, b,
        /*c_mod=*/(short)0, c, /*reuse_a=*/false, /*reuse_b=*/false);
}

#define LOG2E  1.4426950408889634f

// Hardware-transcendental sigmoid: 1/(1+2^(-x*log2e)) -> v_exp_f32 + v_rcp_f32
__device__ __forceinline__ float fast_sigmoid(float x) {
#if __has_builtin(__builtin_amdgcn_exp2f) && __has_builtin(__builtin_amdgcn_rcpf)
    return __builtin_amdgcn_rcpf(1.0f + __builtin_amdgcn_exp2f(-LOG2E * x));
#else
    return 1.0f / (1.0f + __expf(-x));
#endif
}

// Hardware tanh: v_tanh_f32 if the builtin exists, else exp2/rcp formulation.
__device__ __forceinline__ float fast_tanh(float x) {
#if __has_builtin(__builtin_amdgcn_tanhf)
    return __builtin_amdgcn_tanhf(x);
#elif __has_builtin(__builtin_amdgcn_exp2f) && __has_builtin(__builtin_amdgcn_rcpf)
    // tanh(x) = 1 - 2/(2^(2x*log2e) + 1)
    float e = __builtin_amdgcn_exp2f(2.0f * LOG2E * x);
    return 1.0f - 2.0f * __builtin_amdgcn_rcpf(e + 1.0f);
#else
    return tanhf(x);
#endif
}

// One workgroup owns 16 batch rows for the full sequence.
// 8 waves; gate GEMM: wave w computes gate columns [16w, 16w+16).
__global__ __launch_bounds__(256)
void lstm3_fused(const float* __restrict__ x,
                 const float* __restrict__ Wih0, const float* __restrict__ Whh0,
                 const float* __restrict__ bih0, const float* __restrict__ bhh0,
                 const float* __restrict__ Wih1, const float* __restrict__ Whh1,
                 const float* __restrict__ bih1, const float* __restrict__ bhh1,
                 const float* __restrict__ Wih2, const float* __restrict__ Whh2,
                 const float* __restrict__ bih2, const float* __restrict__ bhh2,
                 const float* __restrict__ w1,  const float* __restrict__ b1,
                 const float* __restrict__ w2,  const float* __restrict__ b2,
                 const float* __restrict__ w3,  const float* __restrict__ b3,
                 float* __restrict__ out)
{
    __shared__ float hls[3][16][H_DIM];   // per-layer hidden state
    __shared__ float cls[3][16][H_DIM];   // per-layer cell state
    __shared__ float xpad[16][8];         // layer-0 input tile, K padded 5->8
    __shared__ float gbuf[16][G_DIM];     // gate pre-activations
    __shared__ float t1[16][H_DIM];       // head intermediates
    __shared__ float t2[16][H_DIM];

    const int tid  = threadIdx.x;
    const int lane = tid & 31;
    const int wave = tid >> 5;        // 0..7
    const int half = lane >> 4;       // 0/1: K sub-group for 16x16x4 f32 layout
    const int mn   = lane & 15;       // A row / B col / C col
    const int m0   = blockIdx.x * 16; // batch base for this workgroup

    // zero-init persistent state + pad columns of x tile
    for (int i = tid; i < 3 * 16 * H_DIM; i += 256) {
        ((float*)hls)[i] = 0.f;
        ((float*)cls)[i] = 0.f;
    }
    if (tid < 16 * 8) ((float*)xpad)[tid] = 0.f;
    __syncthreads();

    const float* WihL[3] = {Wih0, Wih1, Wih2};
    const float* WhhL[3] = {Whh0, Whh1, Whh2};
    const float* bihL[3] = {bih0, bih1, bih2};
    const float* bhhL[3] = {bhh0, bhh1, bhh2};

    for (int t = 0; t < S_LEN; ++t) {
        // Stage x[t, m0..m0+15, 0..4] into padded LDS tile (cols 5..7 stay 0).
        // This barrier also orders all cross-step reuse (t1/t2/gbuf/xpad).
        if (tid < 16 * I_DIM) {
            int r = tid / I_DIM, c = tid % I_DIM;
            xpad[r][c] = x[(size_t)t * B_SZ * I_DIM + (size_t)(m0 + r) * I_DIM + c];
        }
        // Prefetch next timestep's x tile (global_prefetch_b8).
        if (t + 1 < S_LEN && tid == 0) {
            __builtin_prefetch(&x[(size_t)(t + 1) * B_SZ * I_DIM + (size_t)m0 * I_DIM], 0, 1);
        }
        __syncthreads();

        #pragma unroll
        for (int L = 0; L < 3; ++L) {
            const int col = wave * 16 + mn;  // gate column this lane produces
            v8f acc = {};

            // ---- input projection: A = x tile (L==0) or h[L-1], B = Wih_L^T ----
            const float* A  = (L == 0) ? &xpad[0][0] : &hls[(L == 0) ? 0 : (L - 1)][0][0];
            const int lda   = (L == 0) ? 8 : H_DIM;
            const int Kin   = (L == 0) ? 8 : H_DIM;
            const int Kreal = (L == 0) ? I_DIM : H_DIM;
            const int ldwi  = (L == 0) ? I_DIM : H_DIM;
            const float* Wi = WihL[L];
            #pragma unroll
            for (int k0 = 0; k0 < Kin; k0 += 4) {
                const int k = k0 + 2 * half;
                v2f a = { A[mn * lda + k], A[mn * lda + k + 1] };
                v2f b = { (k     < Kreal) ? Wi[col * ldwi + k    ] : 0.f,
                          (k + 1 < Kreal) ? Wi[col * ldwi + k + 1] : 0.f };
                acc = wmma4(a, b, acc);
            }

            // ---- recurrent GEMM: A = h[L], B = Whh_L^T ----
            const float* Wh = WhhL[L];
            #pragma unroll
            for (int k0 = 0; k0 < H_DIM; k0 += 4) {
                const int k = k0 + 2 * half;
                v2f a = { hls[L][mn][k], hls[L][mn][k + 1] };
                v2f b = { Wh[col * H_DIM + k], Wh[col * H_DIM + k + 1] };
                acc = wmma4(a, b, acc);
            }

            // biases (bih + bhh, same for every row of the tile), spill to LDS
            const float bsum = bihL[L][col] + bhhL[L][col];
            #pragma unroll
            for (int r = 0; r < 8; ++r)
                gbuf[r + 8 * half][col] = acc[r] + bsum;
            __syncthreads();

            // ---- LSTM cell elementwise: 16x32 cells, 2 per thread ----
            #pragma unroll
            for (int e = 0; e < 2; ++e) {
                const int cell = tid + e * 256;
                const int r = cell >> 5, c = cell & 31;
                const float gi = gbuf[r][c];
                const float gf = gbuf[r][c + 32];
                const float gg = gbuf[r][c + 64];
                const float go = gbuf[r][c + 96];
                const float cn = fast_sigmoid(gf) * cls[L][r][c]
                               + fast_sigmoid(gi) * fast_tanh(gg);
                cls[L][r][c] = cn;
                hls[L][r][c] = fast_sigmoid(go) * fast_tanh(cn);
            }
            __syncthreads();
        }

        // ---- MLP head: t1 = h2@w1^T+b1 ; t2 = t1@w2^T+b2 ; y = relu(t2@w3^T+b3)
        if (wave < 2) {  // wave-uniform branch: active waves keep EXEC all-ones
            const int col = wave * 16 + mn;
            v8f acc = {};
            #pragma unroll
            for (int k0 = 0; k0 < H_DIM; k0 += 4) {
                const int k = k0 + 2 * half;
                v2f a = { hls[2][mn][k], hls[2][mn][k + 1] };
                v2f b = { w1[col * H_DIM + k], w1[col * H_DIM + k + 1] };
                acc = wmma4(a, b, acc);
            }
            const float bb = b1[col];
            #pragma unroll
            for (int r = 0; r < 8; ++r) t1[r + 8 * half][col] = acc[r] + bb;
        }
        __syncthreads();
        if (wave < 2) {
            const int col = wave * 16 + mn;
            v8f acc = {};
            #pragma unroll
            for (int k0 = 0; k0 < H_DIM; k0 += 4) {
                const int k = k0 + 2 * half;
                v2f a = { t1[mn][k], t1[mn][k + 1] };
                v2f b = { w2[col * H_DIM + k], w2[col * H_DIM + k + 1] };
                acc = wmma4(a, b, acc);
            }
            const float bb = b2[col];
            #pragma unroll
            for (int r = 0; r < 8; ++r) t2[r + 8 * half][col] = acc[r] + bb;
        }
        __syncthreads();
        if (tid < 16) {
            float d = b3[0];
            #pragma unroll
            for (int k = 0; k < H_DIM; ++k) d += t2[tid][k] * w3[k];
            out[(size_t)t * B_SZ + m0 + tid] = fmaxf(d, 0.f);
        }
        // No end-of-step barrier needed: next iteration's staging barrier
        // orders every remaining cross-step LDS dependency.
    }
}

extern "C" void kernel_launch(void* const* d_in, const int* in_sizes, int n_in,
                              void* d_out, int out_size, void* d_ws, size_t ws_size,
                              hipStream_t stream) {
    (void)in_sizes; (void)n_in; (void)out_size; (void)d_ws; (void)ws_size;
    const float* x    = (const float*)d_in[0];
    const float* Wih0 = (const float*)d_in[1];
    const float* Whh0 = (const float*)d_in[2];
    const float* bih0 = (const float*)d_in[3];
    const float* bhh0 = (const float*)d_in[4];
    const float* Wih1 = (const float*)d_in[5];
    const float* Whh1 = (const float*)d_in[6];
    const float* bih1 = (const float*)d_in[7];
    const float* bhh1 = (const float*)d_in[8];
    const float* Wih2 = (const float*)d_in[9];
    const float* Whh2 = (const float*)d_in[10];
    const float* bih2 = (const float*)d_in[11];
    const float* bhh2 = (const float*)d_in[12];
    const float* w1   = (const float*)d_in[13];
    const float* b1   = (const float*)d_in[14];
    const float* w2   = (const float*)d_in[15];
    const float* b2   = (const float*)d_in[16];
    const float* w3   = (const float*)d_in[17];
    const float* b3   = (const float*)d_in[18];
    float* out = (float*)d_out;

    dim3 grid(B_SZ / 16);   // 32 persistent workgroups, one 16-row batch slice each
    dim3 block(256);        // 8 waves (wave32)
    lstm3_fused<<<grid, block, 0, stream>>>(x,
                                            Wih0, Whh0, bih0, bhh0,
                                            Wih1, Whh1, bih1, bhh1,
                                            Wih2, Whh2, bih2, bhh2,
                                            w1, b1, w2, b2, w3, b3, out);
}